// AdderDeconv_new_77034533421672
// MI455X (gfx1250) — compile-verified
//
#include <hip/hip_runtime.h>
#include <cstdint>
#include <cstddef>

// ---------------------------------------------------------------------------
// AdderNet decoder for MI455X (gfx1250).
//  * L1-distance "conv" is not bilinear -> no WMMA possible; VALU-bound
//    (2 VALU ops per |x-w| via sub + sub-with-|abs| modifier).
//  * Weights: uniform indices -> scalar loads (SGPR operands, free for VALU).
//  * Activations: staged in LDS, double-buffered, interior filled with the
//    CDNA5 async global->LDS DMA path (global_load_async_to_lds_b32 /
//    s_wait_asynccnt), halo ring with a masked VGPR path.
//  * All dims compile-time -> shifts/masks instead of div/mod.
//  * BN+ReLU fused into producer epilogues; unpool is a scatter kernel.
// ---------------------------------------------------------------------------

#define TH_T 8
#define TW_T 32
#define CI_CHUNK 8
#define LDS_ROW (TW_T + 2)                 // 34 floats
#define LDS_PLANE (LDS_ROW * (TH_T + 2))   // 340 floats per ci
#define LDS_BUF (LDS_PLANE * CI_CHUNK)     // 2720 floats per buffer

__device__ __forceinline__ void async_load_lds_b32(uint32_t lds_byte,
                                                   uint32_t glb_byte_off,
                                                   const float* sbase) {
#if defined(__gfx1250__)
  // vdst = LDS byte address, vaddr = 32-bit global offset, saddr = 64-bit base
  asm volatile("global_load_async_to_lds_b32 %0, %1, %2"
               :: "v"(lds_byte), "v"(glb_byte_off), "s"(sbase)
               : "memory");
#else
  (void)lds_byte; (void)glb_byte_off; (void)sbase;
#endif
}

__device__ __forceinline__ void wait_async0() {
#if defined(__gfx1250__)
#if __has_builtin(__builtin_amdgcn_s_wait_asynccnt)
  __builtin_amdgcn_s_wait_asynccnt(0);
#else
  asm volatile("s_wait_asynccnt 0x0" ::: "memory");
#endif
#endif
}

// ---------------------------------------------------------------------------
// adder layer: out[n,co,h,w] = -sum_{ci,di,dj} |in[n,ci,h+di-1,w+dj-1] - w[co,ci,di,dj]|
// optionally followed by v = relu(v)*bnw[co] + bnb[co]  (fused producer epilogue)
// Grid: x = spatial tiles (W/32 * H/8), y = COUT/NCO, z = N. Block: 256.
// ---------------------------------------------------------------------------
template<int CIN, int COUT, int NCO, bool POST_BN, int H, int W>
__global__ __launch_bounds__(256)
void adder_layer(const float* __restrict__ x, const float* __restrict__ w,
                 const float* __restrict__ bnw, const float* __restrict__ bnb,
                 float* __restrict__ out)
{
  __shared__ float tile[2 * LDS_BUF];

  constexpr int HW = H * W;
  constexpr int TILES_W = W / TW_T;
  constexpr int NCHUNK = CIN / CI_CHUNK;

  const int tid = threadIdx.x;
  const int tw  = tid & 31;
  const int th  = tid >> 5;             // one wave == one tile row (conflict-free LDS)
  const int tx = blockIdx.x % TILES_W;  // compile-time power of two -> mask
  const int ty = blockIdx.x / TILES_W;
  const int w0 = tx * TW_T;
  const int h0 = ty * TH_T;
  const int co0 = blockIdx.y * NCO;
  const int n   = blockIdx.z;

  // workgroup-relative LDS byte address of the tile (flat ptr low 32 bits
  // are the LDS byte address per the gfx1250 aperture layout)
  const uint32_t lds_base = (uint32_t)(uintptr_t)(&tile[0]);

  float acc[NCO];
#pragma unroll
  for (int u = 0; u < NCO; ++u) acc[u] = 0.f;

  // Halo ring mapping: 2*34 + 2*8 = 84 elements per ci plane, threads 0..83.
  int hr = 0, hc = 0;
  const bool hasHalo = (tid < 84);
  if (tid < 34)       { hr = 0;        hc = tid; }
  else if (tid < 68)  { hr = TH_T + 1; hc = tid - 34; }
  else                { int k = tid - 68; hr = 1 + (k >> 1); hc = (k & 1) ? (TW_T + 1) : 0; }

  auto fill = [&](int chunk, int buf) {
    const int cc0 = chunk * CI_CHUNK;
    // ---- interior (rows 1..TH, cols 1..TW): always in-bounds -> async DMA ----
    const float* gbase = x + (size_t)(n * CIN + cc0) * HW + (size_t)h0 * W + w0;
    uint32_t goff = (uint32_t)((th * W + tw) * 4);
    uint32_t loff = lds_base + (uint32_t)((buf * LDS_BUF + (th + 1) * LDS_ROW + (tw + 1)) * 4);
#pragma unroll
    for (int cl = 0; cl < CI_CHUNK; ++cl) {
      async_load_lds_b32(loff, goff, gbase);
      goff += (uint32_t)(HW * 4);
      loff += (uint32_t)(LDS_PLANE * 4);
    }
    // ---- halo ring: masked VGPR path (zero padding at image borders) ----
    if (hasHalo) {
      const int gh = h0 - 1 + hr;
      const int gw = w0 - 1 + hc;
      const bool ok = (gh >= 0) & (gh < H) & (gw >= 0) & (gw < W);
#pragma unroll
      for (int cl = 0; cl < CI_CHUNK; ++cl) {
        float v = 0.f;
        if (ok) v = x[(size_t)(n * CIN + cc0 + cl) * HW + (size_t)gh * W + gw];
        tile[buf * LDS_BUF + cl * LDS_PLANE + hr * LDS_ROW + hc] = v;
      }
    }
  };

  fill(0, 0);
  wait_async0();
  __syncthreads();

  for (int chunk = 0; chunk < NCHUNK; ++chunk) {
    const int buf = chunk & 1;
    if (chunk + 1 < NCHUNK) fill(chunk + 1, buf ^ 1);   // prefetch overlaps compute

    const int cc0 = chunk * CI_CHUNK;
#pragma unroll
    for (int cl = 0; cl < CI_CHUNK; ++cl) {
      const float* tp = &tile[buf * LDS_BUF + cl * LDS_PLANE + th * LDS_ROW + tw];
      float xv[9];
#pragma unroll
      for (int di = 0; di < 3; ++di)
#pragma unroll
        for (int dj = 0; dj < 3; ++dj)
          xv[di * 3 + dj] = tp[di * LDS_ROW + dj];
      // uniform weight addresses -> scalar loads; VALU sees SGPR operands
      const float* wp = w + (size_t)(co0 * CIN + cc0 + cl) * 9;
#pragma unroll
      for (int u = 0; u < NCO; ++u) {
#pragma unroll
        for (int k = 0; k < 9; ++k)
          acc[u] -= fabsf(xv[k] - wp[(size_t)u * CIN * 9 + k]);
      }
    }
    wait_async0();      // retire prefetch issued above (had full compute to hide)
    __syncthreads();
  }

  const int oh = h0 + th, ow = w0 + tw;
#pragma unroll
  for (int u = 0; u < NCO; ++u) {
    const int co = co0 + u;
    float v = acc[u];
    if constexpr (POST_BN) v = fmaxf(v, 0.f) * bnw[co] + bnb[co];
    out[(size_t)(n * COUT + co) * HW + (size_t)oh * W + ow] = v;
  }
}

// ---------------------------------------------------------------------------
// unpool2x + relu + bn:  out zero positions get bn bias b[c] (relu(0)*w+b),
// scattered position gets relu(v)*w[c]+b[c]. Each thread owns one source
// element and writes its private 2x2 output block -> no atomics needed.
// All dims compile-time (powers of two) -> shifts/masks, no div chains.
// ---------------------------------------------------------------------------
template<int C, int Hs, int Ws, int N>
__global__ __launch_bounds__(256)
void unpool_bnrelu(const float* __restrict__ in, const int* __restrict__ idx,
                   const float* __restrict__ bnw, const float* __restrict__ bnb,
                   float* __restrict__ out)
{
  const int i = blockIdx.x * 256 + threadIdx.x;
  int t = i;
  const int ws = t % Ws; t /= Ws;
  const int hs = t % Hs; t /= Hs;
  const int c  = t % C;  const int n = t / C;
  constexpr int Wo = 2 * Ws, Ho = 2 * Hs;

  const float v  = in[i];
  const float bw = bnw[c], bb = bnb[c];
  const float val = fmaxf(v, 0.f) * bw + bb;
  const int target = idx[i];                       // flat index within Ho*Wo
  const size_t obase = (size_t)(n * C + c) * (Ho * Wo);
  const int p00 = (2 * hs) * Wo + 2 * ws;
#pragma unroll
  for (int r = 0; r < 2; ++r)
#pragma unroll
    for (int q = 0; q < 2; ++q) {
      const int p = p00 + r * Wo + q;
      out[obase + p] = (p == target) ? val : bb;
    }
}

// ---------------------------------------------------------------------------
extern "C" void kernel_launch(void* const* d_in, const int* in_sizes, int n_in,
                              void* d_out, int out_size, void* d_ws, size_t ws_size,
                              hipStream_t stream) {
  (void)in_sizes; (void)n_in; (void)out_size; (void)ws_size;

  const float* x      = (const float*)d_in[0];
  const int*   locs13 = (const int*)  d_in[3];
  const int*   locs6  = (const int*)  d_in[4];
  const float* w3  = (const float*)d_in[5];
  const float* w6  = (const float*)d_in[6];
  const float* w9  = (const float*)d_in[7];
  const float* w12 = (const float*)d_in[8];
  const float* w16 = (const float*)d_in[9];
  const float* w19 = (const float*)d_in[10];
  const float* w23 = (const float*)d_in[11];
  const float* w26 = (const float*)d_in[12];
  const float* bn5w  = (const float*)d_in[13]; const float* bn5b  = (const float*)d_in[14];
  const float* bn8w  = (const float*)d_in[15]; const float* bn8b  = (const float*)d_in[16];
  const float* bn11w = (const float*)d_in[17]; const float* bn11b = (const float*)d_in[18];
  const float* bn15w = (const float*)d_in[19]; const float* bn15b = (const float*)d_in[20];
  const float* bn18w = (const float*)d_in[21]; const float* bn18b = (const float*)d_in[22];
  const float* bn22w = (const float*)d_in[23]; const float* bn22b = (const float*)d_in[24];
  const float* bn25w = (const float*)d_in[25]; const float* bn25b = (const float*)d_in[26];

  // ping-pong workspace: two 8 MB halves (max live tensor = 4*32*128*128 f32)
  float* A = (float*)d_ws;
  float* B = A + (8u * 1024u * 1024u) / 4u;
  float* O = (float*)d_out;

  // H=W=32 layers: tiles = 4; NCO=4 -> 512 blocks for latency hiding
  adder_layer<128,128,4,true , 32, 32><<<dim3(4, 32, 4), 256, 0, stream>>>(x, w3,  bn5w,  bn5b,  A);
  adder_layer<128,128,4,true , 32, 32><<<dim3(4, 32, 4), 256, 0, stream>>>(A, w6,  bn8w,  bn8b,  B);
  adder_layer<128,128,4,true , 32, 32><<<dim3(4, 32, 4), 256, 0, stream>>>(B, w9,  bn11w, bn11b, A);
  adder_layer<128, 64,4,false, 32, 32><<<dim3(4, 16, 4), 256, 0, stream>>>(A, w12, nullptr, nullptr, B);

  // unpool (4,64,32,32) -> (4,64,64,64), fused relu+bn15
  unpool_bnrelu<64, 32, 32, 4><<<dim3((4*64*32*32)/256), 256, 0, stream>>>(B, locs13, bn15w, bn15b, A);

  // H=W=64 layers: tiles = 16
  adder_layer< 64, 64,8,true , 64, 64><<<dim3(16, 8, 4), 256, 0, stream>>>(A, w16, bn18w, bn18b, B);
  adder_layer< 64, 32,4,false, 64, 64><<<dim3(16, 8, 4), 256, 0, stream>>>(B, w19, nullptr, nullptr, A);

  // unpool (4,32,64,64) -> (4,32,128,128), fused relu+bn22
  unpool_bnrelu<32, 64, 64, 4><<<dim3((4*32*64*64)/256), 256, 0, stream>>>(A, locs6, bn22w, bn22b, B);

  // H=W=128 layers: tiles = 64
  adder_layer< 32, 32,8,true , 128, 128><<<dim3(64, 4, 4), 256, 0, stream>>>(B, w23, bn25w, bn25b, A);
  adder_layer< 32,  3,3,false, 128, 128><<<dim3(64, 1, 4), 256, 0, stream>>>(A, w26, nullptr, nullptr, O);
}